// WaveNet_36292473651999
// MI455X (gfx1250) — compile-verified
//
#include <hip/hip_runtime.h>

// WaveNet forward for MI455X (gfx1250, wave32, WMMA).
// Every conv is a channel-space GEMM with M = B*T = 131072 positions.
// 32 launches (init + 30 layers + head); launch boundaries are the inter-layer
// global barrier. All matmuls use V_WMMA_F32_16X16X4_F32 (full f32: the gated
// tanh*sigmoid chain over 30 layers is rounding-sensitive).
// Roofline: ~0.95 TFLOP f32 compute, ~12 GB traffic (~0.5 ms @ 23.3 TB/s) ->
// compute-bound on the f32 WMMA pipe; weights (~14 MB) stay L2-resident.
// M-tile 64 per block => each weight fragment load feeds 4 WMMAs; b128
// tap-pair weight loads cover both dilated taps at once. Gate activations run
// on hardware transcendentals (v_exp_f32 / v_rcp_f32) instead of libm tanhf.
//
// Workspace layout (256 MB):
//   [0,   64MB)  : hA  [B,T,128]  position-major h (ping)
//   [64, 128MB)  : hB  [B,T,128]  (pong)
//   [128,256MB)  : skip [B,T,256] skip accumulator (layer 0 writes, rest RMW)

typedef float v2f __attribute__((ext_vector_type(2)));
typedef float v4f __attribute__((ext_vector_type(4)));
typedef float v8f __attribute__((ext_vector_type(8)));

#define IN_CH   256
#define RES_CH  128
#define SKIP_CH 256
#define TLEN    16384
#define BATCH   8
#define NLAYERS 30

__device__ __forceinline__ v8f wmma4_f32(v2f a, v2f b, v8f c) {
  // V_WMMA_F32_16X16X4_F32 : D(16x16,f32) = A(16x4,f32) * B(4x16,f32) + C
  return __builtin_amdgcn_wmma_f32_16x16x4_f32(
      /*neg_a=*/false, a, /*neg_b=*/false, b,
      /*c_mod=*/(short)0, c, /*reuse_a=*/false, /*reuse_b=*/false);
}

// Gate activations on hardware transcendentals (v_exp_f32, v_rcp_f32).
// tanh(x) = 1 - 2/(e^{2x}+1); sigmoid(x) = 1/(1+e^{-x}).
// Both saturate exactly at +/-1 (0/1) for large |x| since exp->{0,inf}.
__device__ __forceinline__ float fast_tanh(float x) {
  return 1.0f - 2.0f * __builtin_amdgcn_rcpf(__expf(2.0f * x) + 1.0f);
}
__device__ __forceinline__ float fast_sigmoid(float x) {
  return __builtin_amdgcn_rcpf(1.0f + __expf(-x));
}

// ---------------------------------------------------------------------------
// Init 1x1 conv: h0[b,t,o] = sum_c Wi[o,c] * x[b,c,t] + bi[o]
// M-tile = 64 (4 sub-tiles/wave): each Wi fragment load feeds 4 WMMAs.
// x is [B,256,T]: for fixed c, lanes 0..15 read consecutive t -> coalesced.
// h0 is [B,T,128] so layers get contiguous channel vectors.
// ---------------------------------------------------------------------------
__global__ void __launch_bounds__(256) wn_init(
    const float* __restrict__ x, const float* __restrict__ Wi,
    const float* __restrict__ bi, float* __restrict__ h0) {
  const int lane = threadIdx.x & 31;
  const int wave = threadIdx.x >> 5;
  const int mrow = lane & 15;          // A: M row; B/C: N col for this lane
  const int hi   = lane >> 4;          // 0: K{0,1}/M v ; 1: K{2,3}/M v+8
  const int koff = hi * 2;

  const size_t p0 = (size_t)blockIdx.x * 64;   // global position base
  const int b  = (int)(p0 / TLEN);
  const int t0 = (int)(p0 % TLEN);

  const int n0 = wave * 16;
  const int oc = n0 + mrow;

  v8f acc[4];
  const float bias = bi[oc];
#pragma unroll
  for (int m = 0; m < 4; ++m)
#pragma unroll
    for (int v = 0; v < 8; ++v) acc[m][v] = bias;

  const float* xb = x + (size_t)b * IN_CH * TLEN + t0;
  const float* wp = Wi + (size_t)oc * IN_CH;
#pragma unroll 2
  for (int k = 0; k < IN_CH; k += 4) {
    const int c0 = k + koff;
    v2f w = *(const v2f*)&wp[c0];              // contiguous -> b64, L2 hit
#pragma unroll
    for (int m = 0; m < 4; ++m) {
      const int t = m * 16 + mrow;
      v2f a;
      a[0] = xb[(size_t)c0 * TLEN + t];        // coalesced across lanes in t
      a[1] = xb[(size_t)(c0 + 1) * TLEN + t];
      acc[m] = wmma4_f32(a, w, acc[m]);
    }
  }

  float* hp = h0 + p0 * RES_CH;
#pragma unroll
  for (int m = 0; m < 4; ++m)
#pragma unroll
    for (int v = 0; v < 8; ++v) {
      const int M = m * 16 + v + hi * 8;
      hp[(size_t)M * RES_CH + oc] = acc[m][v];
    }
}

// ---------------------------------------------------------------------------
// One residual layer, M-tile = 64 positions per block.
// Stage 1 (wave w -> output channels 16w..16w+15 of BOTH f and g):
//   loop over 128 channels in steps of 4; one b128 load gives both taps'
//   weight fragments; per M sub-tile two b64 A loads (h[t-d], h[t]) feed
//   4 WMMAs (f/g x tap0/tap1). Causal pad: clamped address + per-lane select
//   (EXEC stays all-ones at every WMMA).
// Gate z = tanh(f)*sigmoid(g) -> LDS (hardware exp/rcp).
// Stage 2: 24 n-tiles over K=128 from LDS (16 skip + 8 residual), 3 per wave,
//   each weight fragment feeds 4 WMMAs (4 M sub-tiles).
// ---------------------------------------------------------------------------
__global__ void __launch_bounds__(256) wn_layer(
    const float* __restrict__ h_in, float* __restrict__ h_out,
    float* __restrict__ skip,
    const float* __restrict__ Wf, const float* __restrict__ bf,
    const float* __restrict__ Wg, const float* __restrict__ bg,
    const float* __restrict__ Wr, const float* __restrict__ br,
    const float* __restrict__ Ws, const float* __restrict__ bs,
    const int dilation, const int accumulate) {
  __shared__ float zsm[64][RES_CH + 2];        // stride 130 floats (8B-aligned)

  const int lane = threadIdx.x & 31;
  const int wave = threadIdx.x >> 5;
  const int mrow = lane & 15;
  const int hi   = lane >> 4;
  const int koff = hi * 2;

  const size_t p0 = (size_t)blockIdx.x * 64;
  const int b  = (int)(p0 / TLEN);
  const int t0 = (int)(p0 % TLEN);

  const int n0 = wave * 16;
  const int oc = n0 + mrow;

  // ---- stage 1: f and g ---------------------------------------------------
  v8f accF[4], accG[4];
  const float bF = bf[oc], bG = bg[oc];
#pragma unroll
  for (int m = 0; m < 4; ++m)
#pragma unroll
    for (int v = 0; v < 8; ++v) { accF[m][v] = bF; accG[m][v] = bG; }

  const float* hb = h_in + (size_t)b * TLEN * RES_CH;
  const float* a1p[4];                         // tap1: h[t]
  const float* a0p[4];                         // tap0: h[t-d] (clamped addr)
  bool zf[4];                                  // tap0 is causal zero-pad?
#pragma unroll
  for (int m = 0; m < 4; ++m) {
    const int t  = t0 + m * 16 + mrow;
    const int tm = t - dilation;
    zf[m]  = (tm < 0);
    a1p[m] = hb + (size_t)t * RES_CH;
    a0p[m] = hb + (size_t)(tm < 0 ? 0 : tm) * RES_CH;
  }

  const float* wfp = Wf + (size_t)oc * (RES_CH * 2);   // [c][tap], stride 2
  const float* wgp = Wg + (size_t)oc * (RES_CH * 2);
#pragma unroll 2
  for (int c = 0; c < RES_CH; c += 4) {
    const int c0 = c + koff;
    // one b128 per matrix covers channels {c0,c0+1} x taps {0,1}
    v4f wf4 = *(const v4f*)&wfp[c0 * 2];
    v4f wg4 = *(const v4f*)&wgp[c0 * 2];
    v2f wf0, wf1, wg0, wg1;
    wf0[0] = wf4[0]; wf0[1] = wf4[2];  wf1[0] = wf4[1]; wf1[1] = wf4[3];
    wg0[0] = wg4[0]; wg0[1] = wg4[2];  wg1[0] = wg4[1]; wg1[1] = wg4[3];
#pragma unroll
    for (int m = 0; m < 4; ++m) {
      v2f a0 = *(const v2f*)&a0p[m][c0];
      a0[0] = zf[m] ? 0.0f : a0[0];            // per-lane select, EXEC intact
      a0[1] = zf[m] ? 0.0f : a0[1];
      v2f a1 = *(const v2f*)&a1p[m][c0];
      accF[m] = wmma4_f32(a0, wf0, accF[m]);
      accF[m] = wmma4_f32(a1, wf1, accF[m]);
      accG[m] = wmma4_f32(a0, wg0, accG[m]);
      accG[m] = wmma4_f32(a1, wg1, accG[m]);
    }
  }

  // gate -> LDS
#pragma unroll
  for (int m = 0; m < 4; ++m)
#pragma unroll
    for (int v = 0; v < 8; ++v) {
      const float z = fast_tanh(accF[m][v]) * fast_sigmoid(accG[m][v]);
      zsm[m * 16 + v + hi * 8][oc] = z;
    }
  __syncthreads();

  // ---- stage 2: skip (tiles 0..15) + residual (tiles 16..23) --------------
  for (int tile = wave * 3; tile < wave * 3 + 3; ++tile) {  // uniform in wave
    const bool is_res = (tile >= 16);
    const int n2  = (is_res ? (tile - 16) : tile) * 16;
    const int oc2 = n2 + mrow;
    const float* Wp  = is_res ? (Wr + (size_t)oc2 * RES_CH)
                              : (Ws + (size_t)oc2 * RES_CH);
    const float bias = is_res ? br[oc2] : bs[oc2];
    v8f acc[4];
#pragma unroll
    for (int m = 0; m < 4; ++m)
#pragma unroll
      for (int v = 0; v < 8; ++v) acc[m][v] = bias;

#pragma unroll 4
    for (int k = 0; k < RES_CH; k += 4) {
      const int c0 = k + koff;
      v2f w = *(const v2f*)&Wp[c0];            // b64
#pragma unroll
      for (int m = 0; m < 4; ++m) {
        v2f a = *(const v2f*)&zsm[m * 16 + mrow][c0];   // ds_load_b64
        acc[m] = wmma4_f32(a, w, acc[m]);
      }
    }

#pragma unroll
    for (int m = 0; m < 4; ++m)
#pragma unroll
      for (int v = 0; v < 8; ++v) {
        const int M = m * 16 + v + hi * 8;
        const size_t row = p0 + M;             // == b*T + (t0+M)
        if (is_res) {
          h_out[row * RES_CH + oc2] = h_in[row * RES_CH + oc2] + acc[m][v];
        } else {
          float* sp = &skip[row * SKIP_CH + oc2];
          *sp = accumulate ? (*sp + acc[m][v]) : acc[m][v];
        }
      }
  }
}

// ---------------------------------------------------------------------------
// Head: s = relu(skip); s2 = relu(W1@s + b1); out = W2@s2 + b2.
// M-tile = 32 (keeps the s2 LDS staging buffer ~33 KB). k-outer loops share
// each A fragment between both n-tiles of a wave. relu fused into A loads;
// output written in the reference's [B,256,T] layout.
// ---------------------------------------------------------------------------
__global__ void __launch_bounds__(256) wn_head(
    const float* __restrict__ skip,
    const float* __restrict__ W1, const float* __restrict__ b1,
    const float* __restrict__ W2, const float* __restrict__ b2,
    float* __restrict__ out) {
  __shared__ float ssm[32][SKIP_CH + 2];       // stride 258 floats

  const int lane = threadIdx.x & 31;
  const int wave = threadIdx.x >> 5;
  const int mrow = lane & 15;
  const int hi   = lane >> 4;
  const int koff = hi * 2;

  const size_t p0 = (size_t)blockIdx.x * 32;
  const int b  = (int)(p0 / TLEN);
  const int t0 = (int)(p0 % TLEN);

  const int oc0 = (wave * 2 + 0) * 16 + mrow;  // n-tile 0 column
  const int oc1 = (wave * 2 + 1) * 16 + mrow;  // n-tile 1 column

  // ---- GEMM1: relu(skip) -> s2, N=256 (A shared across both n-tiles) ------
  {
    const float* wp0 = W1 + (size_t)oc0 * SKIP_CH;
    const float* wp1 = W1 + (size_t)oc1 * SKIP_CH;
    v8f acc[2][2];                             // [n-tile][m]
    const float b0 = b1[oc0], b1v = b1[oc1];
#pragma unroll
    for (int m = 0; m < 2; ++m)
#pragma unroll
      for (int v = 0; v < 8; ++v) { acc[0][m][v] = b0; acc[1][m][v] = b1v; }

#pragma unroll 2
    for (int k = 0; k < SKIP_CH; k += 4) {
      const int c0 = k + koff;
      v2f w0 = *(const v2f*)&wp0[c0];
      v2f w1 = *(const v2f*)&wp1[c0];
#pragma unroll
      for (int m = 0; m < 2; ++m) {
        v2f a = *(const v2f*)&skip[(p0 + m * 16 + mrow) * SKIP_CH + c0];
        a[0] = fmaxf(a[0], 0.0f);              // relu(skip) fused
        a[1] = fmaxf(a[1], 0.0f);
        acc[0][m] = wmma4_f32(a, w0, acc[0][m]);
        acc[1][m] = wmma4_f32(a, w1, acc[1][m]);
      }
    }
#pragma unroll
    for (int m = 0; m < 2; ++m)
#pragma unroll
      for (int v = 0; v < 8; ++v) {
        const int r = m * 16 + v + hi * 8;
        ssm[r][oc0] = fmaxf(acc[0][m][v], 0.0f);
        ssm[r][oc1] = fmaxf(acc[1][m][v], 0.0f);
      }
  }
  __syncthreads();

  // ---- GEMM2: out = W2 @ s2 + b2, N=256 -----------------------------------
  {
    const float* wp0 = W2 + (size_t)oc0 * SKIP_CH;
    const float* wp1 = W2 + (size_t)oc1 * SKIP_CH;
    v8f acc[2][2];
    const float b0 = b2[oc0], b1v = b2[oc1];
#pragma unroll
    for (int m = 0; m < 2; ++m)
#pragma unroll
      for (int v = 0; v < 8; ++v) { acc[0][m][v] = b0; acc[1][m][v] = b1v; }

#pragma unroll 2
    for (int k = 0; k < SKIP_CH; k += 4) {
      const int c0 = k + koff;
      v2f w0 = *(const v2f*)&wp0[c0];
      v2f w1 = *(const v2f*)&wp1[c0];
#pragma unroll
      for (int m = 0; m < 2; ++m) {
        v2f a = *(const v2f*)&ssm[m * 16 + mrow][c0];
        acc[0][m] = wmma4_f32(a, w0, acc[0][m]);
        acc[1][m] = wmma4_f32(a, w1, acc[1][m]);
      }
    }
#pragma unroll
    for (int m = 0; m < 2; ++m)
#pragma unroll
      for (int v = 0; v < 8; ++v) {
        const int M = m * 16 + v + hi * 8;
        out[((size_t)b * IN_CH + oc0) * TLEN + (t0 + M)] = acc[0][m][v];
        out[((size_t)b * IN_CH + oc1) * TLEN + (t0 + M)] = acc[1][m][v];
      }
  }
}

// ---------------------------------------------------------------------------
extern "C" void kernel_launch(void* const* d_in, const int* in_sizes, int n_in,
                              void* d_out, int out_size, void* d_ws, size_t ws_size,
                              hipStream_t stream) {
  (void)in_sizes; (void)n_in; (void)out_size; (void)ws_size;
  const float* x  = (const float*)d_in[0];
  const float* Wi = (const float*)d_in[1];
  const float* bi = (const float*)d_in[2];
  const float* Wf = (const float*)d_in[3];
  const float* bf = (const float*)d_in[4];
  const float* Wg = (const float*)d_in[5];
  const float* bg = (const float*)d_in[6];
  const float* Wr = (const float*)d_in[7];
  const float* br = (const float*)d_in[8];
  const float* Ws = (const float*)d_in[9];
  const float* bs = (const float*)d_in[10];
  const float* W1 = (const float*)d_in[11];
  const float* b1 = (const float*)d_in[12];
  const float* W2 = (const float*)d_in[13];
  const float* b2 = (const float*)d_in[14];
  float* out = (float*)d_out;

  char* ws = (char*)d_ws;
  const size_t H_BYTES = (size_t)BATCH * TLEN * RES_CH * sizeof(float);   // 64 MB
  float* hA = (float*)(ws);
  float* hB = (float*)(ws + H_BYTES);
  float* sk = (float*)(ws + 2 * H_BYTES);                                 // 128 MB

  const int P = BATCH * TLEN;             // 131072 positions

  wn_init<<<P / 64, 256, 0, stream>>>(x, Wi, bi, hA);

  const float* hin = hA;
  float* hout = hB;
  for (int i = 0; i < NLAYERS; ++i) {
    const int d = 1 << (i % 10);
    wn_layer<<<P / 64, 256, 0, stream>>>(
        hin, hout, sk,
        Wf + (size_t)i * RES_CH * RES_CH * 2, bf + (size_t)i * RES_CH,
        Wg + (size_t)i * RES_CH * RES_CH * 2, bg + (size_t)i * RES_CH,
        Wr + (size_t)i * RES_CH * RES_CH,     br + (size_t)i * RES_CH,
        Ws + (size_t)i * SKIP_CH * RES_CH,    bs + (size_t)i * SKIP_CH,
        d, (i == 0) ? 0 : 1);
    const float* tmp = hout; hout = (float*)hin; hin = tmp;
  }

  wn_head<<<P / 32, 256, 0, stream>>>(sk, W1, b1, W2, b2, out);
}